// LambadaRankLoss_790273982468
// MI455X (gfx1250) — compile-verified
//
#include <hip/hip_runtime.h>
#include <math.h>

#define N_ITEMS 8192
#define CUTOFF  10
#define NT      1024          // one workgroup = 32 wave32s on a single WGP
#define NWAVES  (NT / 32)

__device__ __forceinline__ float softplus_f(float x) {
    // stable: max(x,0) + log1p(exp(-|x|))
    float ax = fabsf(x);
    return fmaxf(x, 0.0f) + log1pf(__expf(-ax));
}

// 1/log2(k+2) for k = 0..9 (NDCG discounts / idcg weights)
__device__ __constant__ float DISC_TAB[CUTOFF] = {
    1.0f,          0.6309297536f, 0.5f,          0.4306765581f, 0.3868528072f,
    0.3562071871f, 0.3333333333f, 0.3154648768f, 0.3010299957f, 0.2890648263f
};

// wave32 argmax butterfly: larger value wins, ties -> smaller index (stable)
__device__ __forceinline__ void wave_argmax(float& v, int& i) {
#pragma unroll
    for (int m = 16; m > 0; m >>= 1) {
        float v2 = __shfl_xor(v, m, 32);
        int   i2 = __shfl_xor(i, m, 32);
        if (v2 > v || (v2 == v && i2 < i)) { v = v2; i = i2; }
    }
}

// block-wide argmax over NT threads; winner returned to ALL threads.
// Caller must place a __syncthreads() before swv/swi are written again.
__device__ __forceinline__ void block_argmax(float& v, int& i,
                                             float* swv, int* swi, int tid) {
    wave_argmax(v, i);
    const int wave = tid >> 5, lane = tid & 31;
    if (lane == 0) { swv[wave] = v; swi[wave] = i; }
    __syncthreads();
    if (wave == 0) {                    // exactly 32 waves -> one wave finishes
        v = swv[lane]; i = swi[lane];
        wave_argmax(v, i);
        if (lane == 0) { swv[0] = v; swi[0] = i; }
    }
    __syncthreads();
    v = swv[0]; i = swi[0];
}

__global__ __launch_bounds__(NT)
void lrl_fused_kernel(const float* __restrict__ o,
                      const float* __restrict__ s,
                      float* __restrict__ out) {
    __shared__ __align__(16) float sh_o[N_ITEMS];   // 32 KB
    __shared__ __align__(16) float sh_s[N_ITEMS];   // 32 KB
    __shared__ float swv[NWAVES];
    __shared__ int   swi[NWAVES];
    __shared__ int   l_i[CUTOFF];                   // top-o indices
    __shared__ float l_o[CUTOFF], l_s[CUTOFF], l_g[CUTOFF];
    __shared__ int   l_i2[CUTOFF];                  // top-s indices
    __shared__ float l_sb[CUTOFF];                  // top-s values

    const int tid = threadIdx.x;

    // ---- stage both arrays into LDS (CDNA5 async copy path, ASYNCcnt) ----
#if defined(__gfx1250__)
    {
        // 1024 lanes x 16 B x 2 chunks per array = 32 KB per array.
        unsigned lo0 = (unsigned)(size_t)&sh_o[tid * 4];
        unsigned lo1 = (unsigned)(size_t)&sh_o[tid * 4 + 4096];
        unsigned ls0 = (unsigned)(size_t)&sh_s[tid * 4];
        unsigned ls1 = (unsigned)(size_t)&sh_s[tid * 4 + 4096];
        unsigned long long go0 = (unsigned long long)(size_t)(o + tid * 4);
        unsigned long long go1 = (unsigned long long)(size_t)(o + tid * 4 + 4096);
        unsigned long long gs0 = (unsigned long long)(size_t)(s + tid * 4);
        unsigned long long gs1 = (unsigned long long)(size_t)(s + tid * 4 + 4096);
        asm volatile("global_load_async_to_lds_b128 %0, %1, off" :: "v"(lo0), "v"(go0) : "memory");
        asm volatile("global_load_async_to_lds_b128 %0, %1, off" :: "v"(lo1), "v"(go1) : "memory");
        asm volatile("global_load_async_to_lds_b128 %0, %1, off" :: "v"(ls0), "v"(gs0) : "memory");
        asm volatile("global_load_async_to_lds_b128 %0, %1, off" :: "v"(ls1), "v"(gs1) : "memory");
        asm volatile("s_wait_asynccnt 0" ::: "memory");
    }
#else
    for (int i = tid; i < N_ITEMS; i += NT) { sh_o[i] = o[i]; sh_s[i] = s[i]; }
#endif
    __syncthreads();

    // ---- phase A: top-CUTOFF of o; destroy winners in LDS, no exclusion ----
    for (int it = 0; it < CUTOFF; ++it) {
        float bv = -INFINITY; int bi = 0x7fffffff;
        for (int j = tid; j < N_ITEMS; j += NT) {
            float v = sh_o[j];
            if (v > bv || (v == bv && j < bi)) { bv = v; bi = j; }
        }
        block_argmax(bv, bi, swv, swi, tid);
        if (tid == 0) {
            l_i[it] = bi; l_o[it] = bv; l_s[it] = sh_s[bi];
            sh_o[bi] = -INFINITY;              // remove from later rounds
        }
        __syncthreads();
    }
    if (tid < CUTOFF) {
        sh_o[l_i[tid]] = l_o[tid];             // restore originals for phase C
        l_g[tid] = exp2f(l_s[tid]) - 1.0f;     // gains of top-o items
    }
    __syncthreads();

    // ---- phase B: top-CUTOFF of s -> idcg (identical in all threads) ----
    float idcg = 0.0f;
    for (int it = 0; it < CUTOFF; ++it) {
        float bv = -INFINITY; int bi = 0x7fffffff;
        for (int j = tid; j < N_ITEMS; j += NT) {
            float v = sh_s[j];
            if (v > bv || (v == bv && j < bi)) { bv = v; bi = j; }
        }
        block_argmax(bv, bi, swv, swi, tid);
        idcg += (exp2f(bv) - 1.0f) * DISC_TAB[it];
        if (tid == 0) {
            l_i2[it] = bi; l_sb[it] = bv;
            sh_s[bi] = -INFINITY;
        }
        __syncthreads();
    }
    if (tid < CUTOFF) sh_s[l_i2[tid]] = l_sb[tid];   // restore
    __syncthreads();
    const float inv_idcg = 1.0f / idcg;

    // ---- phase C: pair sweep, both ordered directions per (top_k, j) ----
    float acc = 0.0f;
    for (int j = tid; j < N_ITEMS; j += NT) {
        const float oj = sh_o[j];
        const float sj = sh_s[j];
        const float gj = exp2f(sj) - 1.0f;
        float dj = 0.0f;                       // disc_j: nonzero only in top-10
#pragma unroll
        for (int k = 0; k < CUTOFF; ++k)
            if (l_i[k] == j) dj = DISC_TAB[k];
#pragma unroll
        for (int k = 0; k < CUTOFF; ++k) {
            const float oa = l_o[k];
            if (oa == oj) continue;            // reference zeroes equal-o pairs
            const float delta = fabsf((l_g[k] - gj) * (DISC_TAB[k] - dj)) * inv_idcg;
            const float diff  = oa - oj;       // o_a - o_j
            const float sa    = l_s[k];
            const float bce_ab = (sa > sj) ? softplus_f(-diff) : softplus_f(diff);
            const float bce_ba = (sj > sa) ? softplus_f(diff)  : softplus_f(-diff);
            acc += (bce_ab + bce_ba) * delta;
        }
    }
    // Ordered top x top pairs were counted twice above: subtract each once.
    if (tid < CUTOFF * CUTOFF) {
        const int a = tid / CUTOFF, b = tid % CUTOFF;
        if (a != b && l_o[a] != l_o[b]) {
            const float delta = fabsf((l_g[a] - l_g[b]) *
                                      (DISC_TAB[a] - DISC_TAB[b])) * inv_idcg;
            const float diff  = l_o[a] - l_o[b];
            const float bce   = (l_s[a] > l_s[b]) ? softplus_f(-diff)
                                                  : softplus_f(diff);
            acc -= bce * delta;
        }
    }

    // ---- deterministic block sum: wave butterflies + one LDS hop ----
#pragma unroll
    for (int m = 16; m > 0; m >>= 1) acc += __shfl_xor(acc, m, 32);
    const int wave = tid >> 5, lane = tid & 31;
    __syncthreads();                            // swv free for reuse
    if (lane == 0) swv[wave] = acc;
    __syncthreads();
    if (wave == 0) {
        acc = swv[lane];
#pragma unroll
        for (int m = 16; m > 0; m >>= 1) acc += __shfl_xor(acc, m, 32);
        if (lane == 0) out[0] = acc / (float)N_ITEMS;
    }
}

extern "C" void kernel_launch(void* const* d_in, const int* in_sizes, int n_in,
                              void* d_out, int out_size, void* d_ws, size_t ws_size,
                              hipStream_t stream) {
    const float* o = (const float*)d_in[0];   // outputs (8192,1) f32
    const float* s = (const float*)d_in[1];   // scores  (8192,1) f32
    (void)d_ws; (void)ws_size; (void)in_sizes; (void)n_in; (void)out_size;

    lrl_fused_kernel<<<1, NT, 0, stream>>>(o, s, (float*)d_out);
}